// SSGC_31980326486700
// MI455X (gfx1250) — compile-verified
//
#include <hip/hip_runtime.h>

#define HID   128
#define OUTD  64
#define NLAYERS 4
#define ALPHA_C 0.6f

typedef float v2f __attribute__((ext_vector_type(2)));
typedef float v8f __attribute__((ext_vector_type(8)));

// ---------------------------------------------------------------------------
// small elementwise kernels
// ---------------------------------------------------------------------------
__global__ void k_fill_f32(float* __restrict__ p, float v, int n) {
  int i = blockIdx.x * blockDim.x + threadIdx.x;
  if (i < n) p[i] = v;
}

__global__ void k_zero_f4(float4* __restrict__ p, int n4) {
  int i = blockIdx.x * blockDim.x + threadIdx.x;
  if (i < n4) p[i] = make_float4(0.f, 0.f, 0.f, 0.f);
}

__global__ void k_deg_count(const long long* __restrict__ dst,
                            float* __restrict__ deg, int E) {
  int i = blockIdx.x * blockDim.x + threadIdx.x;
  if (i < E) atomicAdd(&deg[dst[i]], 1.0f);
}

__global__ void k_rsqrt_inplace(float* __restrict__ p, int n) {
  int i = blockIdx.x * blockDim.x + threadIdx.x;
  if (i < n) p[i] = rsqrtf(p[i]);
}

// ---------------------------------------------------------------------------
// fp32 WMMA GEMM:  C[M,NC] = A[M,K] @ B[K,NC] (+ bias), V_WMMA_F32_16X16X4_F32
//
//  - one wave computes TWO 16-row tiles (rows tile0*16.. and tile1*16..), so
//    every B fragment ds_load_b64 feeds two WMMAs (16 wmma : 8 ds : 2 vmem / k-step)
//  - B staged in LDS pair-interleaved: pair-row p holds (B[2p][c], B[2p+1][c])
//    adjacent -> each fragment is ONE aligned ds_load_b64 into the WMMA src pair
//    (no repacking movs). Odd pair-rows are column-XOR-16 swizzled so the two
//    half-waves (pair-rows p vs p+1) hit disjoint LDS bank halves.
//  - A fragment load software-pipelined one k-step ahead.
//  - MIX variant fuses A = c1*A + c2*A2 into the fragment load (residual mix).
//
//  Fragment layouts per ISA:
//   A (16x4 f32): lanes 0-15 rows M0..15 K{k,k+1}; lanes 16-31 K{k+2,k+3}
//   B (4x16 f32): VGPR0 = row k+2*lh, VGPR1 = row k+1+2*lh, lane col = l15
//   C/D (16x16):  VGPR j -> row j (lanes 0-15) / row j+8 (lanes 16-31)
//  Requires M % 16 == 0 at tile granularity (N=100000 ok) and K <= 128.
// ---------------------------------------------------------------------------
template <int NT, bool MIX>   // NT = NC/16 (8 -> NC=128, 4 -> NC=64)
__global__ __launch_bounds__(256) void k_gemm_wmma(
    const float* __restrict__ A, const float* __restrict__ A2,
    float c1, float c2,
    const float* __restrict__ B, const float* __restrict__ bias,
    float* __restrict__ C, int M, int K) {
  constexpr int NC = NT * 16;
  __shared__ float ldsB[64 * NC * 2];   // K_max/2 = 64 pair-rows, interleaved

  // ---- stage B: interleave row pairs, XOR-16 column swizzle on odd pairs ----
  {
    const int tasks = (K >> 1) * (NC >> 2);
    for (int t = threadIdx.x; t < tasks; t += 256) {
      const int pr = t / (NC >> 2);
      const int c4 = (t - pr * (NC >> 2)) << 2;
      const float4 lo = *(const float4*)(B + (size_t)(2 * pr) * NC + c4);
      const float4 hi = *(const float4*)(B + (size_t)(2 * pr + 1) * NC + c4);
      const int cs = c4 ^ ((pr & 1) << 4);
      v2f* dst = (v2f*)&ldsB[(pr * NC + cs) * 2];
      v2f p0 = {lo.x, hi.x}; v2f p1 = {lo.y, hi.y};
      v2f p2 = {lo.z, hi.z}; v2f p3 = {lo.w, hi.w};
      dst[0] = p0; dst[1] = p1; dst[2] = p2; dst[3] = p3;
    }
  }
  __syncthreads();

  const int wave = threadIdx.x >> 5;
  const int lane = threadIdx.x & 31;
  const int l15  = lane & 15;
  const int lh   = lane >> 4;

  const int tile0 = blockIdx.x * 16 + wave;   // block covers 16 row tiles
  const int tile1 = tile0 + 8;
  const int row0  = tile0 * 16;
  const int row1  = tile1 * 16;
  if (row0 >= M) return;                      // no barriers after this point
  const bool valid1 = (row1 < M);

  const float* Ap0 = A + (size_t)(row0 + l15) * K + 2 * lh;
  const float* Ap1 = A + (size_t)((valid1 ? row1 : row0) + l15) * K + 2 * lh;
  const float* Aq0 = MIX ? (A2 + (size_t)(row0 + l15) * K + 2 * lh) : Ap0;
  const float* Aq1 = MIX ? (A2 + (size_t)((valid1 ? row1 : row0) + l15) * K + 2 * lh) : Ap1;

  auto loadA = [&](const float* p, const float* q, int k) -> v2f {
    v2f a = *(const v2f*)(p + k);
    if (MIX) {
      v2f a2 = *(const v2f*)(q + k);
      a = a * c1 + a2 * c2;
    }
    return a;
  };

  v8f acc0[NT], acc1[NT];
#pragma unroll
  for (int t = 0; t < NT; ++t) {
    acc0[t] = (v8f){0.f,0.f,0.f,0.f,0.f,0.f,0.f,0.f};
    acc1[t] = (v8f){0.f,0.f,0.f,0.f,0.f,0.f,0.f,0.f};
  }

  // B fragment address: ldsB[(p+lh)*NC*2 + ((t^lh)*16 + l15)*2], p = k/2 (even)
  const float* bbase = ldsB + (lh * NC + l15) * 2;
  const int dlh = lh << 5;

  v2f a0 = loadA(Ap0, Aq0, 0);
  v2f a1 = loadA(Ap1, Aq1, 0);

  for (int k = 0; k < K; k += 4) {
    const v2f ca0 = a0, ca1 = a1;
    if (k + 4 < K) {                          // software pipeline next A frag
      a0 = loadA(Ap0, Aq0, k + 4);
      a1 = loadA(Ap1, Aq1, k + 4);
    }
    const float* pe = bbase + dlh;            // even t: +lh pair offset
    const float* po = bbase - dlh;            // odd  t: -lh (XOR-16 swizzle)
#pragma unroll
    for (int t = 0; t < NT; ++t) {
      const float* bp = ((t & 1) ? po : pe) + (t << 5);
      const v2f b = *(const v2f*)bp;          // single aligned ds_load_b64
      acc0[t] = __builtin_amdgcn_wmma_f32_16x16x4_f32(
          false, ca0, false, b, (short)0, acc0[t], false, false);
      acc1[t] = __builtin_amdgcn_wmma_f32_16x16x4_f32(
          false, ca1, false, b, (short)0, acc1[t], false, false);
    }
    bbase += NC * 4;                          // p advances by 2 per k-step
  }

#pragma unroll
  for (int t = 0; t < NT; ++t) {
    const int col = t * 16 + l15;
    const float bv = bias ? bias[col] : 0.0f;
#pragma unroll
    for (int j = 0; j < 8; ++j) {
      C[(size_t)(row0 + j + 8 * lh) * NC + col] = acc0[t][j] + bv;
      if (valid1)
        C[(size_t)(row1 + j + 8 * lh) * NC + col] = acc1[t][j] + bv;
    }
  }
}

// ---------------------------------------------------------------------------
// edge scatter: wave per edge, float4 per lane; grid-stride loop with
// prefetch of the next edge's source row (global_prefetch_b8).
//   agg[dst] += XW[src] * (dinv[src]*dinv[dst])
// XW (51.2 MB) lives in the 192 MB L2, so gathers are L2-bandwidth bound.
// ---------------------------------------------------------------------------
__global__ __launch_bounds__(256) void k_scatter(
    const long long* __restrict__ ei, const float* __restrict__ XW,
    const float* __restrict__ dinv, float* __restrict__ agg, int E) {
  const int wave = threadIdx.x >> 5;
  const int lane = threadIdx.x & 31;
  const long long stride = (long long)gridDim.x * 8;
  for (long long e = (long long)blockIdx.x * 8 + wave; e < E; e += stride) {
    const long long s = ei[e];
    const long long d = ei[E + e];
    const long long en = e + stride;
    if (en < E) {                       // warm L2->L0 path for next gather
      const long long sn = ei[en];
      __builtin_prefetch(XW + sn * (long long)HID + (lane << 2), 0, 3);
    }
    const float norm = dinv[s] * dinv[d];
    const float4 v = ((const float4*)(XW + s * (long long)HID))[lane];
    float* ap = agg + d * (long long)HID + (lane << 2);
    atomicAdd(ap + 0, v.x * norm);
    atomicAdd(ap + 1, v.y * norm);
    atomicAdd(ap + 2, v.z * norm);
    atomicAdd(ap + 3, v.w * norm);
  }
}

// ---------------------------------------------------------------------------
// fused layer epilogue: Z = relu(agg + XW*dinv^2 + b); z_sum += Z   (float4)
// ---------------------------------------------------------------------------
__global__ __launch_bounds__(256) void k_finalize(
    const float4* __restrict__ agg, const float4* __restrict__ XW,
    const float* __restrict__ dinv, const float* __restrict__ bias,
    float4* __restrict__ Z, float4* __restrict__ zsum, int n4) {
  int i = blockIdx.x * blockDim.x + threadIdx.x;
  if (i >= n4) return;
  const int n = i >> 5;                 // HID/4 = 32 float4 per row
  const int q = i & 31;
  const float di = dinv[n];
  const float sl = di * di;
  const float4 bb = ((const float4*)bias)[q];
  const float4 a = agg[i];
  const float4 x = XW[i];
  float4 v;
  v.x = fmaxf(fmaf(x.x, sl, a.x) + bb.x, 0.f);
  v.y = fmaxf(fmaf(x.y, sl, a.y) + bb.y, 0.f);
  v.z = fmaxf(fmaf(x.z, sl, a.z) + bb.z, 0.f);
  v.w = fmaxf(fmaf(x.w, sl, a.w) + bb.w, 0.f);
  Z[i] = v;
  float4 zs = zsum[i];
  zs.x += v.x; zs.y += v.y; zs.z += v.z; zs.w += v.w;
  zsum[i] = zs;
}

// ---------------------------------------------------------------------------
extern "C" void kernel_launch(void* const* d_in, const int* in_sizes, int n_in,
                              void* d_out, int out_size, void* d_ws, size_t ws_size,
                              hipStream_t stream) {
  const float*     x     = (const float*)d_in[0];
  const long long* ei    = (const long long*)d_in[1];   // int64 edge_index [2,E]
  const float*     W1    = (const float*)d_in[2];
  const float*     b1    = (const float*)d_in[3];
  const float*     convW = (const float*)d_in[4];
  const float*     convB = (const float*)d_in[5];
  const float*     W2    = (const float*)d_in[6];
  const float*     b2    = (const float*)d_in[7];
  float*           out   = (float*)d_out;

  const int N = in_sizes[0] / HID;   // IN_DIM == HID == 128
  const int E = in_sizes[1] / 2;

  // ---- workspace carve-up ----
  char* ws = (char*)d_ws;
  float* dinv = (float*)ws;
  size_t off = ((size_t)N * sizeof(float) + 255) & ~(size_t)255;
  const size_t big = (size_t)N * HID * sizeof(float);
  float* Zinit = (float*)(ws + off);            off += big;
  float* Z     = (float*)(ws + off);            off += big;
  float* XW    = (float*)(ws + off);            off += big;
  float* agg   = (float*)(ws + off);            off += big;
  float* zsum  = (float*)(ws + off);            off += big;

  const int n4   = N * (HID / 4);
  const int gN   = (N + 255) / 256;
  const int gE   = (E + 255) / 256;
  const int g4   = (n4 + 255) / 256;
  const int tiles = (N + 15) / 16;
  const int gGemm = (tiles + 15) / 16;          // 16 row tiles per block
  const int gScat = (E + 63) / 64;              // ~8 edges per wave

  // 1) symmetric-norm degrees: deg = 1 + segment_sum(1, dst); dinv = rsqrt(deg)
  k_fill_f32<<<gN, 256, 0, stream>>>(dinv, 1.0f, N);
  k_deg_count<<<gE, 256, 0, stream>>>(ei + E, dinv, E);
  k_rsqrt_inplace<<<gN, 256, 0, stream>>>(dinv, N);

  // 2) Z_init = x @ W1 + b1
  k_gemm_wmma<8, false><<<gGemm, 256, 0, stream>>>(x, x, 0.f, 0.f,
                                                   W1, b1, Zinit, N, HID);

  // 3) z_sum = 0
  k_zero_f4<<<g4, 256, 0, stream>>>((float4*)zsum, n4);

  // 4) layers
  for (int i = 0; i < NLAYERS; ++i) {
    const float* Zin = (i == 0) ? Zinit : Z;
    k_gemm_wmma<8, false><<<gGemm, 256, 0, stream>>>(
        Zin, Zin, 0.f, 0.f, convW + (size_t)i * HID * HID, nullptr, XW, N, HID);
    k_zero_f4<<<g4, 256, 0, stream>>>((float4*)agg, n4);
    k_scatter<<<gScat, 256, 0, stream>>>(ei, XW, dinv, agg, E);
    k_finalize<<<g4, 256, 0, stream>>>((const float4*)agg, (const float4*)XW,
                                       dinv, convB + (size_t)i * HID,
                                       (float4*)Z, (float4*)zsum, n4);
  }

  // 5) out = (ALPHA*z_init + ((1-ALPHA)/L)*z_sum) @ W2 + b2  (mix fused in A load)
  k_gemm_wmma<4, true><<<gGemm, 256, 0, stream>>>(
      Zinit, zsum, ALPHA_C, (1.0f - ALPHA_C) / (float)NLAYERS,
      W2, b2, out, N, HID);
}